// RouterV2_65068754534649
// MI455X (gfx1250) — compile-verified
//
#include <hip/hip_runtime.h>
#include <hip/hip_bf16.h>
#include <math.h>

typedef __attribute__((ext_vector_type(2))) float v2f;
typedef __attribute__((ext_vector_type(8))) float v8f;

#define RB 4
#define RT 8192
#define RD 2048
#define RCAP 4096

// ---------------------------------------------------------------------------
// Kernel 1: logits = x @ w via V_WMMA_F32_16X16X4_F32 (fp32, matches ref).
// One wave -> 16 tokens. blockDim = 256 (8 waves) -> 128 tokens per block.
// A fragment (16x4 f32): lanes 0-15 hold K=0,1 ; lanes 16-31 hold K=2,3 for
// row M = lane&15  => per-lane contiguous float2 load from the token row.
// B fragment (4x16): w[k..k+3] replicated across N (same slot->K mapping),
// so every column of D equals the dot product; read column 0 (lanes 0 / 16).
// ---------------------------------------------------------------------------
__global__ __launch_bounds__(256) void router_logits_wmma_kernel(
    const float* __restrict__ x, const float* __restrict__ w,
    float* __restrict__ logits) {
  const int wave = threadIdx.x >> 5;
  const int lane = threadIdx.x & 31;
  const int tok0 = (blockIdx.x * 8 + wave) * 16;
  const int row  = lane & 15;        // token within the 16-token group
  const int koff = (lane >> 4) << 1; // 0 for lanes 0-15, 2 for lanes 16-31

  const float* __restrict__ xrow = x + (size_t)(tok0 + row) * RD + koff;
  const float* __restrict__ wk   = w + koff;

  v8f c = {0.f, 0.f, 0.f, 0.f, 0.f, 0.f, 0.f, 0.f};

#pragma unroll 8
  for (int k = 0; k < RD; k += 4) {
    v2f a = *(const v2f*)(xrow + k);   // A[M][k+koff .. k+koff+1]
    v2f b = *(const v2f*)(wk + k);     // B[k+koff .. k+koff+1] (bcast over N)
    c = __builtin_amdgcn_wmma_f32_16x16x4_f32(
        /*neg_a=*/false, a, /*neg_b=*/false, b,
        /*c_mod=*/(short)0, c, /*reuse_a=*/false, /*reuse_b=*/false);
  }

  // C/D 16x16 f32 layout: VGPR r -> lanes 0-15: M=r, N=lane ; lanes 16-31:
  // M=8+r, N=lane-16.  Column N=0 lives in lane 0 (M=0..7) and lane 16 (M=8..15).
  if (lane == 0 || lane == 16) {
    const int base = tok0 + (lane >> 4) * 8;
#pragma unroll
    for (int r = 0; r < 8; ++r) logits[base + r] = c[r];
  }
}

// ---------------------------------------------------------------------------
// Block reduction helper (blockDim must be power of two).
// ---------------------------------------------------------------------------
__device__ __forceinline__ float block_reduce_sum(float v, float* sred) {
  const int tid = threadIdx.x, nt = blockDim.x;
  sred[tid] = v;
  __syncthreads();
  for (int s = nt >> 1; s > 0; s >>= 1) {
    if (tid < s) sred[tid] += sred[tid + s];
    __syncthreads();
  }
  float r = sred[0];
  __syncthreads();
  return r;
}

// ---------------------------------------------------------------------------
// Kernel 2: per batch row -- bitonic sort (desc by value, asc by index, i.e.
// the jax.lax.top_k order), top-4096 indices, masked softmax scatter, aux
// partials. One 1024-thread workgroup per batch; 52KB LDS (<< 320KB WGP).
// ---------------------------------------------------------------------------
__global__ __launch_bounds__(1024) void router_topk_softmax_kernel(
    const float* __restrict__ logits, float* __restrict__ out_w,
    float* __restrict__ out_idx, float* __restrict__ partials) {
  __shared__ float          sv[RT];
  __shared__ unsigned short si[RT];
  __shared__ float          sred[1024];

  const int b   = blockIdx.x;
  const int tid = threadIdx.x;
  const int nt  = blockDim.x;
  const float* __restrict__ lg = logits + (size_t)b * RT;

  for (int i = tid; i < RT; i += nt) {
    sv[i] = lg[i];
    si[i] = (unsigned short)i;
  }
  __syncthreads();

  // Bitonic sort, ascending under strict order: (v desc, idx asc).
  for (int k = 2; k <= RT; k <<= 1) {
    for (int j = k >> 1; j > 0; j >>= 1) {
      for (int t = tid; t < RT; t += nt) {
        const int ixj = t ^ j;
        if (ixj > t) {
          const float v0 = sv[t], v1 = sv[ixj];
          const int   i0 = si[t], i1 = si[ixj];
          // less(a,b) := a.v > b.v || (a.v == b.v && a.i < b.i)
          const bool less_ixj_t = (v1 > v0) || (v1 == v0 && i1 < i0);
          const bool dir_asc    = ((t & k) == 0);
          const bool do_swap    = dir_asc ? less_ixj_t : !less_ixj_t;
          if (do_swap) {
            sv[t] = v1; sv[ixj] = v0;
            si[t] = (unsigned short)i1; si[ixj] = (unsigned short)i0;
          }
        }
      }
      __syncthreads();
    }
  }

  // Top-k indices (top_k order == sorted order here), emitted as float.
  for (int t = tid; t < RCAP; t += nt)
    out_idx[(size_t)b * RCAP + t] = (float)si[t];

  // Zero the routing-weight row, then scatter after a barrier.
  for (int t = tid; t < RT; t += nt) out_w[(size_t)b * RT + t] = 0.0f;
  __threadfence_block();
  __syncthreads();

  const float m = sv[0];  // global max (descending sort)

  float sel_sum = 0.f, all_sum = 0.f, sig_sum = 0.f;
  for (int t = tid; t < RT; t += nt) {
    const float e = __expf(sv[t] - m);
    all_sum += e;
    if (t < RCAP) sel_sum += e;
    sig_sum += 1.0f / (1.0f + __expf(-sv[t]));
  }
  sel_sum = block_reduce_sum(sel_sum, sred);
  all_sum = block_reduce_sum(all_sum, sred);
  sig_sum = block_reduce_sum(sig_sum, sred);

  const float inv_sel = 1.0f / sel_sum;
  for (int t = tid; t < RCAP; t += nt)
    out_w[(size_t)b * RT + si[t]] = __expf(sv[t] - m) * inv_sel;

  if (tid == 0) {
    const float lse = m + __logf(all_sum);
    const float mp  = sig_sum * (1.0f / (float)RT);
    partials[2 * b + 0] = (mp - 0.5f) * (mp - 0.5f);
    partials[2 * b + 1] = lse * lse;
  }
}

// ---------------------------------------------------------------------------
// Kernel 3: fold 4 per-batch partials into the scalar aux loss.
// ---------------------------------------------------------------------------
__global__ void router_finalize_kernel(const float* __restrict__ partials,
                                       float* __restrict__ out_aux) {
  if (threadIdx.x == 0 && blockIdx.x == 0) {
    float lb = 0.f, zz = 0.f;
    for (int i = 0; i < RB; ++i) {
      lb += partials[2 * i + 0];
      zz += partials[2 * i + 1];
    }
    lb *= (1.0f / RB);
    zz *= (1.0f / RB);
    out_aux[0] = 0.01f * lb + 0.001f * zz;
  }
}

extern "C" void kernel_launch(void* const* d_in, const int* in_sizes, int n_in,
                              void* d_out, int out_size, void* d_ws, size_t ws_size,
                              hipStream_t stream) {
  (void)in_sizes; (void)n_in; (void)out_size; (void)ws_size;
  const float* x = (const float*)d_in[0];   // (B,T,D) fp32
  const float* w = (const float*)d_in[1];   // (D,)   fp32
  float* out = (float*)d_out;

  float* logits   = (float*)d_ws;           // B*T floats = 128 KB
  float* partials = logits + (size_t)RB * RT;  // 8 floats

  float* out_w   = out;                          // (B,T,1) weights
  float* out_idx = out + (size_t)RB * RT;        // (B,CAP) indices (as float)
  float* out_aux = out_idx + (size_t)RB * RCAP;  // scalar

  // 32768 tokens / (8 waves * 16 tokens) = 256 blocks of 256 threads.
  router_logits_wmma_kernel<<<dim3(256), dim3(256), 0, stream>>>(x, w, logits);
  router_topk_softmax_kernel<<<dim3(RB), dim3(1024), 0, stream>>>(logits, out_w,
                                                                  out_idx, partials);
  router_finalize_kernel<<<dim3(1), dim3(32), 0, stream>>>(partials, out_aux);
}